// CustomGAT_21981642621100
// MI455X (gfx1250) — compile-verified
//
#include <hip/hip_runtime.h>

typedef __attribute__((ext_vector_type(2))) float v2f;
typedef __attribute__((ext_vector_type(8))) float v8f;

// ---------------------------------------------------------------------------
// WMMA fp32 GEMM:  Y[N,M] = X[N,K] @ W[K,M] + bias[M]   (optional ReLU)
//
// One block (8 wave32s) owns one 16-row tile of Y and all M/16 column tiles
// (M <= 128 here, so <= 8 column tiles -> one per wave). Per 64-deep K chunk,
// the block cooperatively stages the 16xKB X tile and the KBxM W chunk into
// LDS, then each wave runs a fully-unrolled V_WMMA_F32_16X16X4_F32 chain off
// LDS (4 WMMAs per 16-deep group, batched ds_loads with immediate offsets).
//
// Fragment layouts per CDNA5 ISA 7.12.2:
//   A (16x4):  lane = half*16+l16 -> row l16, K pair half*2 (+0/+1 in v[0]/v[1])
//   B (4x16):  rows kb+{half*2, half*2+1}, col l16
//   C/D:       VGPR r -> row r + 8*half, col l16
// Requires N%16==0, M%16==0 (M<=128), K%16==0 — true for all call sites.
// ---------------------------------------------------------------------------
#define KB 64
__global__ void gemm16_wmma(const float* __restrict__ X, const float* __restrict__ W,
                            const float* __restrict__ bias, float* __restrict__ Y,
                            int N, int K, int M, int relu)
{
    __shared__ float Wl[KB * 128];   // <= 32 KB
    __shared__ float Xl[16 * KB];    //    4 KB

    const int r0   = blockIdx.x << 4;         // row tile base
    const int lane = threadIdx.x & 31;
    const int wave = threadIdx.x >> 5;
    const int half = lane >> 4;               // 0 or 1
    const int l16  = lane & 15;
    const int mt   = M >> 4;                  // column tiles (<= 8)

    v8f acc = {};

    for (int kb = 0; kb < K; kb += KB) {
        const int kn = (K - kb < KB) ? (K - kb) : KB;   // 16 or 64, multiple of 16

        // stage W rows [kb, kb+kn): contiguous kn*M floats
        for (int i = threadIdx.x; i < kn * M; i += blockDim.x)
            Wl[i] = W[(size_t)kb * M + i];
        // stage X tile rows [r0, r0+16), cols [kb, kb+kn)
        for (int i = threadIdx.x; i < 16 * kn; i += blockDim.x) {
            int row = i / kn, col = i - row * kn;
            Xl[row * KB + col] = X[(size_t)(r0 + row) * K + kb + col];
        }
        // prefetch next W chunk into cache while we compute (global_prefetch_b8)
        if (kb + KB < K && threadIdx.x < 64)
            __builtin_prefetch(&W[(size_t)(kb + KB) * M + threadIdx.x * 32], 0, 3);
        __syncthreads();

        for (int ct = wave; ct < mt; ct += 8) {
            const int c0 = ct << 4;
            const float* xrow = Xl + l16 * KB + half * 2;
            const float* wcol = Wl + (half * 2) * M + c0 + l16;
            v8f a8 = acc;     // accumulator per column tile (mt<=8 -> 1 tile/wave)
            for (int k = 0; k < kn; k += 16) {
                #pragma unroll
                for (int kk = 0; kk < 16; kk += 4) {   // 4 WMMAs, batched LDS loads
                    const int kc = k + kk;
                    v2f a, b;
                    a.x = xrow[kc];
                    a.y = xrow[kc + 1];
                    b.x = wcol[kc * M];
                    b.y = wcol[kc * M + M];
                    a8 = __builtin_amdgcn_wmma_f32_16x16x4_f32(
                             false, a, false, b, (short)0, a8, false, false);
                }
            }
            acc = a8;
        }
        __syncthreads();
    }

    // store (each wave owns column tile `wave` when wave < mt)
    if (wave < mt) {
        const int c0 = wave << 4;
        float bb = bias[c0 + l16];
        #pragma unroll
        for (int r = 0; r < 8; ++r) {
            int rr = r0 + r + half * 8;
            float v = acc[r] + bb;
            if (relu) v = fmaxf(v, 0.0f);
            Y[(size_t)rr * M + c0 + l16] = v;
        }
    }
}

// ---------------------------------------------------------------------------
// Small helpers
// ---------------------------------------------------------------------------
__device__ __forceinline__ float lrelu02(float v) { return v > 0.0f ? v : 0.2f * v; }

// Float atomic max via sign-split (init slots to -0.0f = 0x80000000).
// Result is >= true max (exact when max >= 0, -0.0 when all-negative), which is
// a valid softmax shift (softmax is shift-invariant; no overflow either way).
__device__ __forceinline__ void atomicMaxF(float* addr, float v)
{
    if (v >= 0.0f) atomicMax((int*)addr, __float_as_int(v));
    else           atomicMin((unsigned int*)addr, (unsigned int)__float_as_int(v));
}

__global__ void fill_u32(unsigned int* __restrict__ p, unsigned int v, long n)
{
    long i = (long)blockIdx.x * blockDim.x + threadIdx.x;
    if (i < n) p[i] = v;
}

// out[i*M + j] = bias[j]  (accumulator init = layer bias)
__global__ void row_bias(float* __restrict__ out, const float* __restrict__ bias,
                         long N, int M)
{
    long i = (long)blockIdx.x * blockDim.x + threadIdx.x;
    if (i < N * M) out[i] = bias[i % M];
}

// y[i] = dot(X[i,:K], w) + b[0] (+ add[i])   -- M=1 heads
__global__ void gemv1(const float* __restrict__ X, const float* __restrict__ w,
                      const float* __restrict__ b, const float* __restrict__ add,
                      float* __restrict__ y, int N, int K)
{
    int i = blockIdx.x * blockDim.x + threadIdx.x;
    if (i >= N) return;
    const float* xr = X + (size_t)i * K;
    float s = 0.0f;
    for (int k = 0; k < K; ++k) s += xr[k] * w[k];
    s += b[0];
    if (add) s += add[i];
    y[i] = s;
}

// ---------------------------------------------------------------------------
// GATv2 edge kernels. Node feature rows are 128 wide = 2 heads x 64 channels.
// One thread per (edge, head). hl/hr/accumulators are L2-resident (51.2 MB
// each vs 192 MB L2), so the gathers and f32 atomics stay on-die.
// ---------------------------------------------------------------------------
__global__ void edge_score(const int* __restrict__ esrc, const int* __restrict__ edst,
                           const float* __restrict__ hl, const float* __restrict__ hr,
                           const float* __restrict__ att,
                           float* __restrict__ score, float* __restrict__ smax, int E)
{
    int t = blockIdx.x * blockDim.x + threadIdx.x;
    if (t >= E * 2) return;
    int e = t >> 1, h = t & 1;
    int s = esrc[e], d = edst[e];
    const float4* pl = (const float4*)(hl + (size_t)s * 128 + h * 64);
    const float4* pr = (const float4*)(hr + (size_t)d * 128 + h * 64);
    const float4* pa = (const float4*)(att + h * 64);
    float sum = 0.0f;
    #pragma unroll 4
    for (int i = 0; i < 16; ++i) {
        float4 a = pl[i], b = pr[i], w = pa[i];
        sum += lrelu02(a.x + b.x) * w.x;
        sum += lrelu02(a.y + b.y) * w.y;
        sum += lrelu02(a.z + b.z) * w.z;
        sum += lrelu02(a.w + b.w) * w.w;
    }
    score[t] = sum;
    atomicMaxF(&smax[d * 2 + h], sum);
}

__global__ void edge_exp(const int* __restrict__ edst,
                         float* __restrict__ score, const float* __restrict__ smax,
                         float* __restrict__ den, int E)
{
    int t = blockIdx.x * blockDim.x + threadIdx.x;
    if (t >= E * 2) return;
    int e = t >> 1, h = t & 1;
    int d = edst[e];
    float ex = __expf(score[t] - smax[d * 2 + h]);
    score[t] = ex;                          // reuse buffer: now holds exp(score - max)
    atomicAdd(&den[d * 2 + h], ex);
}

__global__ void edge_aggr(const int* __restrict__ esrc, const int* __restrict__ edst,
                          const float* __restrict__ hl, const float* __restrict__ ex,
                          const float* __restrict__ den, float* __restrict__ out, int E)
{
    int t = blockIdx.x * blockDim.x + threadIdx.x;
    if (t >= E * 2) return;
    int e = t >> 1, h = t & 1;
    int s = esrc[e], d = edst[e];
    float w = ex[t] / (den[d * 2 + h] + 1e-16f);
    const float4* src = (const float4*)(hl + (size_t)s * 128 + h * 64);
    float* dst = out + (size_t)d * 128 + h * 64;
    #pragma unroll 4
    for (int i = 0; i < 16; ++i) {
        float4 v = src[i];
        atomicAdd(&dst[i * 4 + 0], v.x * w);
        atomicAdd(&dst[i * 4 + 1], v.y * w);
        atomicAdd(&dst[i * 4 + 2], v.z * w);
        atomicAdd(&dst[i * 4 + 3], v.w * w);
    }
}

// ---------------------------------------------------------------------------
// Host-side orchestration
// ---------------------------------------------------------------------------
extern "C" void kernel_launch(void* const* d_in, const int* in_sizes, int n_in,
                              void* d_out, int out_size, void* d_ws, size_t ws_size,
                              hipStream_t stream)
{
    const float* x_pano  = (const float*)d_in[0];
    const float* x_fp    = (const float*)d_in[1];
    const int*   epp_src = (const int*)d_in[2];
    const int*   epp_dst = (const int*)d_in[3];
    const int*   epf_src = (const int*)d_in[4];
    const int*   epf_dst = (const int*)d_in[5];
    const float *c0_Wl = (const float*)d_in[6],  *c0_bl = (const float*)d_in[7];
    const float *c0_Wr = (const float*)d_in[8],  *c0_br = (const float*)d_in[9];
    const float *c0_at = (const float*)d_in[10], *c0_b  = (const float*)d_in[11];
    const float *c1_Wl = (const float*)d_in[12], *c1_bl = (const float*)d_in[13];
    const float *c1_Wr = (const float*)d_in[14], *c1_br = (const float*)d_in[15];
    const float *c1_at = (const float*)d_in[16], *c1_b  = (const float*)d_in[17];
    const float *ct_Wl = (const float*)d_in[18], *ct_bl = (const float*)d_in[19];
    const float *ct_Wr = (const float*)d_in[20], *ct_br = (const float*)d_in[21];
    const float *ct_at = (const float*)d_in[22], *ct_b  = (const float*)d_in[23];
    const float *nm_lw = (const float*)d_in[24], *nm_lb = (const float*)d_in[25];
    const float *nm_sw = (const float*)d_in[26], *nm_sb = (const float*)d_in[27];
    const float *nm_bw = (const float*)d_in[28], *nm_bb = (const float*)d_in[29];
    const float *nm_cw = (const float*)d_in[30], *nm_cb = (const float*)d_in[31];
    const float *m_w1  = (const float*)d_in[32], *m_b1  = (const float*)d_in[33];
    const float *m_w2  = (const float*)d_in[34], *m_b2  = (const float*)d_in[35];
    const float *m_w3  = (const float*)d_in[36], *m_b3  = (const float*)d_in[37];

    const int Np  = in_sizes[0] / 64;   // 100000 (= 16 * 6250)
    const int Nf  = in_sizes[1] / 16;   // 20000  (= 16 * 1250)
    const int Epp = in_sizes[2];        // 1,000,000
    const int Epf = in_sizes[4];        // 200,000

    // -------- workspace layout (floats) --------
    float* ws = (float*)d_ws;
    float* A    = ws;                               // [Np,128] hl        51.2 MB
    float* B    = A    + (size_t)Np * 128;          // [Np,128] hr        51.2 MB
    float* C    = B    + (size_t)Np * 128;          // [Np,128] layer out 51.2 MB
    float* F    = C    + (size_t)Np * 128;          // [Nf,128] fp out    10.2 MB
    float* SC   = F    + (size_t)Nf * 128;          // [Epp*2]             8 MB
    float* SMAX = SC   + (size_t)Epp * 2;           // [Np*2]
    float* DEN  = SMAX + (size_t)Np * 2;            // [Np*2]
    float* NH1  = DEN  + (size_t)Np * 2;            // [Nf,64]
    float* NH2  = NH1  + (size_t)Nf * 64;           // [Nf,64]
    float* LIN  = NH2  + (size_t)Nf * 64;           // [Nf]
    float* LIN2 = LIN  + (size_t)Nf;                // [Nf]

    const int TB = 256;
    #define GRID1(n) dim3((unsigned)(((long)(n) + TB - 1) / TB))
    #define GEMMG(N) dim3((unsigned)((N) >> 4))

    // -------- NullModel on footprint features --------
    gemm16_wmma<<<GEMMG(Nf), TB, 0, stream>>>(x_fp, nm_sw, nm_sb, NH1, Nf, 16, 64, 1);
    gemm16_wmma<<<GEMMG(Nf), TB, 0, stream>>>(NH1, nm_bw, nm_bb, NH2, Nf, 64, 64, 1);
    gemm16_wmma<<<GEMMG(Nf), TB, 0, stream>>>(NH2, nm_bw, nm_bb, NH1, Nf, 64, 64, 1);
    gemv1<<<GRID1(Nf), TB, 0, stream>>>(x_fp, nm_lw, nm_lb, nullptr, LIN, Nf, 16);
    gemv1<<<GRID1(Nf), TB, 0, stream>>>(NH1, nm_cw, nm_cb, LIN, LIN2, Nf, 64);

    // -------- GATv2 layer 0 (pano graph) --------
    gemm16_wmma<<<GEMMG(Np), TB, 0, stream>>>(x_pano, c0_Wl, c0_bl, A, Np, 64, 128, 0);
    gemm16_wmma<<<GEMMG(Np), TB, 0, stream>>>(x_pano, c0_Wr, c0_br, B, Np, 64, 128, 0);
    fill_u32<<<GRID1(Np * 2), TB, 0, stream>>>((unsigned*)SMAX, 0x80000000u, (long)Np * 2);
    fill_u32<<<GRID1(Np * 2), TB, 0, stream>>>((unsigned*)DEN, 0u, (long)Np * 2);
    edge_score<<<GRID1(Epp * 2), TB, 0, stream>>>(epp_src, epp_dst, A, B, c0_at, SC, SMAX, Epp);
    edge_exp<<<GRID1(Epp * 2), TB, 0, stream>>>(epp_dst, SC, SMAX, DEN, Epp);
    row_bias<<<GRID1((long)Np * 128), TB, 0, stream>>>(C, c0_b, Np, 128);
    edge_aggr<<<GRID1(Epp * 2), TB, 0, stream>>>(epp_src, epp_dst, A, SC, DEN, C, Epp);

    // -------- GATv2 layer 1 (pano graph) --------
    gemm16_wmma<<<GEMMG(Np), TB, 0, stream>>>(C, c1_Wl, c1_bl, A, Np, 128, 128, 0);
    gemm16_wmma<<<GEMMG(Np), TB, 0, stream>>>(C, c1_Wr, c1_br, B, Np, 128, 128, 0);
    fill_u32<<<GRID1(Np * 2), TB, 0, stream>>>((unsigned*)SMAX, 0x80000000u, (long)Np * 2);
    fill_u32<<<GRID1(Np * 2), TB, 0, stream>>>((unsigned*)DEN, 0u, (long)Np * 2);
    edge_score<<<GRID1(Epp * 2), TB, 0, stream>>>(epp_src, epp_dst, A, B, c1_at, SC, SMAX, Epp);
    edge_exp<<<GRID1(Epp * 2), TB, 0, stream>>>(epp_dst, SC, SMAX, DEN, Epp);
    row_bias<<<GRID1((long)Np * 128), TB, 0, stream>>>(C, c1_b, Np, 128);
    edge_aggr<<<GRID1(Epp * 2), TB, 0, stream>>>(epp_src, epp_dst, A, SC, DEN, C, Epp);

    // -------- translate conv (pano -> footprint, bipartite) --------
    gemm16_wmma<<<GEMMG(Np), TB, 0, stream>>>(C, ct_Wl, ct_bl, A, Np, 128, 128, 0);
    gemm16_wmma<<<GEMMG(Nf), TB, 0, stream>>>(x_fp, ct_Wr, ct_br, B, Nf, 16, 128, 0);
    fill_u32<<<GRID1(Nf * 2), TB, 0, stream>>>((unsigned*)SMAX, 0x80000000u, (long)Nf * 2);
    fill_u32<<<GRID1(Nf * 2), TB, 0, stream>>>((unsigned*)DEN, 0u, (long)Nf * 2);
    edge_score<<<GRID1(Epf * 2), TB, 0, stream>>>(epf_src, epf_dst, A, B, ct_at, SC, SMAX, Epf);
    edge_exp<<<GRID1(Epf * 2), TB, 0, stream>>>(epf_dst, SC, SMAX, DEN, Epf);
    row_bias<<<GRID1((long)Nf * 128), TB, 0, stream>>>(F, ct_b, Nf, 128);
    edge_aggr<<<GRID1(Epf * 2), TB, 0, stream>>>(epf_src, epf_dst, A, SC, DEN, F, Epf);

    // -------- closing MLP + combine with NullModel skip --------
    gemm16_wmma<<<GEMMG(Nf), TB, 0, stream>>>(F, m_w1, m_b1, NH1, Nf, 128, 64, 1);
    gemm16_wmma<<<GEMMG(Nf), TB, 0, stream>>>(NH1, m_w2, m_b2, NH2, Nf, 64, 64, 1);
    gemv1<<<GRID1(Nf), TB, 0, stream>>>(NH2, m_w3, m_b3, LIN2, (float*)d_out, Nf, 64);

    #undef GRID1
    #undef GEMMG
}